// EstimatorQNNGen104_65481071409448
// MI455X (gfx1250) — compile-verified
//
#include <hip/hip_runtime.h>

typedef __attribute__((ext_vector_type(2))) float v2f;
typedef __attribute__((ext_vector_type(4))) float v4f;
typedef __attribute__((ext_vector_type(8))) float v8f;

#define LOG2E_F 1.44269504088896340736f

__device__ __forceinline__ float fast_sigmoid(float x) {
  return __builtin_amdgcn_rcpf(1.0f + __builtin_amdgcn_exp2f(-x * LOG2E_F));
}

__device__ __forceinline__ float fast_tanh(float x) {
#if __has_builtin(__builtin_amdgcn_tanhf)
  return __builtin_amdgcn_tanhf(x);
#else
  float e2 = __builtin_amdgcn_exp2f(x * (2.0f * LOG2E_F));
  return 1.0f - 2.0f * __builtin_amdgcn_rcpf(e2 + 1.0f);
#endif
}

// Half-swap across the two 16-lane rows: result[L] = v[L ^ 16].
// v_permlanex16_b32 with identity nibble selects — pure VALU, no DS pipe.
__device__ __forceinline__ float half_swap(float v) {
#if __has_builtin(__builtin_amdgcn_permlanex16)
  int i = __float_as_int(v);
  int r = __builtin_amdgcn_permlanex16(i, i, 0x76543210, 0xFEDCBA98, false, false);
  return __int_as_float(r);
#else
  return __shfl_xor(v, 16, 32);
#endif
}

__device__ __forceinline__ v2f mkv2(float a, float b) { v2f r; r.x = a; r.y = b; return r; }

// D = A(16x4 f32) * B(4x16 f32) + C, wave32 WMMA
__device__ __forceinline__ v8f wmma4(v2f a, v2f b, v8f c) {
  return __builtin_amdgcn_wmma_f32_16x16x4_f32(false, a, false, b, (short)0, c, false, false);
}

__global__ __launch_bounds__(256)
void estimator_qnn_kernel(const float* __restrict__ in,
                          const float* __restrict__ conv_w, const float* __restrict__ conv_b,
                          const float* __restrict__ sW1, const float* __restrict__ sb1,
                          const float* __restrict__ sW2, const float* __restrict__ sb2,
                          const float* __restrict__ eW1, const float* __restrict__ eb1,
                          const float* __restrict__ eW2, const float* __restrict__ eb2,
                          const float* __restrict__ eW3, const float* __restrict__ eb3,
                          float* __restrict__ out, int n) {
  const int lane = threadIdx.x & 31;
  const int j    = lane & 15;          // M index within tile
  const int j8   = j & 7;
  const int j4   = j & 3;
  const bool lo  = lane < 16;
  const int kb0  = lo ? 0 : 2;         // K base for this lane-half (32-bit A/B layout)

  // ---- uniform scalar weights ----
  const float cw0 = conv_w[0], cw1 = conv_w[1], cw2 = conv_w[2], cw3 = conv_w[3];
  const float cb  = conv_b[0];
  float s1w0[4], s1w1[4], s1b[4], s2w0[4], s2w1[4];
#pragma unroll
  for (int q = 0; q < 4; ++q) {
    s1w0[q] = sW1[q];        // sW1[0][q]
    s1w1[q] = sW1[4 + q];    // sW1[1][q]
    s1b[q]  = sb1[q];
    s2w0[q] = sW2[2 * q];    // sW2[q][0]
    s2w1[q] = sW2[2 * q + 1];
  }
  const float s2b0 = sb2[0], s2b1 = sb2[1];
  float eb1v[8];
#pragma unroll
  for (int q = 0; q < 8; ++q) eb1v[q] = eb1[q];
  float eb2v[4], w3v[4];
#pragma unroll
  for (int q = 0; q < 4; ++q) { eb2v[q] = eb2[q]; w3v[q] = eW3[q]; }
  const float eb3s = eb3[0];

  // ---- per-lane A operands: A = W^T zero-padded to 16 x K (loop-invariant) ----
  // Layer 1: A1[M=j][k] = eW1[k*8 + j], valid j<8, k<7. Chunk c: k = 4c + kb0 + {0,1}.
  auto ldW1 = [&](int k) -> float {
    float v = eW1[(k < 7 ? k : 6) * 8 + j8];      // clamped, always in-bounds
    return (j < 8 && k < 7) ? v : 0.0f;
  };
  const v2f a1c0 = mkv2(ldW1(kb0 + 0), ldW1(kb0 + 1));
  const v2f a1c1 = mkv2(ldW1(kb0 + 4), ldW1(kb0 + 5));
  // Layer 2: A2[M=j][k] = eW2[k*4 + j], valid j<4, k<8.
  auto ldW2 = [&](int k) -> float {
    float v = eW2[k * 4 + j4];
    return (j < 4) ? v : 0.0f;
  };
  const v2f a2c0 = mkv2(ldW2(kb0 + 0), ldW2(kb0 + 1));
  const v2f a2c1 = mkv2(ldW2(kb0 + 4), ldW2(kb0 + 5));

  // B-register builders (1 permlanex16 + 1 cndmask each).
  // bt0: source values live in lanes 0-15 -> high lanes take the swapped khi value.
  auto bt0 = [&](float vlo, float vhi) -> float {
    float t = half_swap(vhi);
    return lo ? vlo : t;
  };
  // bt1: source values live in lanes 16-31 -> low lanes take the swapped klo value.
  auto bt1 = [&](float vlo, float vhi) -> float {
    float t = half_swap(vlo);
    return lo ? t : vhi;
  };

  const v4f* __restrict__ in4 = (const v4f*)in;
  const int groups  = n >> 5;
  const int waveId  = (blockIdx.x * blockDim.x + threadIdx.x) >> 5;
  const int nWaves  = (gridDim.x * blockDim.x) >> 5;

  for (int g = waveId; g < groups; g += nWaves) {
    const int base = g << 5;
    const v4f x = __builtin_nontemporal_load(&in4[base + lane]);  // coalesced 512B/wave, NT

    // ---- conv (dot4 + sigmoid) ----
    float cacc = fmaf(x.x, cw0, fmaf(x.y, cw1, fmaf(x.z, cw2, fmaf(x.w, cw3, cb))));
    const float fconv = fast_sigmoid(cacc);

    // ---- sampler: Linear(2,4)-tanh-Linear(4,2)-softmax ----
    float t0 = fast_tanh(fmaf(x.x, s1w0[0], fmaf(x.y, s1w1[0], s1b[0])));
    float t1 = fast_tanh(fmaf(x.x, s1w0[1], fmaf(x.y, s1w1[1], s1b[1])));
    float t2 = fast_tanh(fmaf(x.x, s1w0[2], fmaf(x.y, s1w1[2], s1b[2])));
    float t3 = fast_tanh(fmaf(x.x, s1w0[3], fmaf(x.y, s1w1[3], s1b[3])));
    float z0 = fmaf(t0, s2w0[0], fmaf(t1, s2w0[1], fmaf(t2, s2w0[2], fmaf(t3, s2w0[3], s2b0))));
    float z1 = fmaf(t0, s2w1[0], fmaf(t1, s2w1[1], fmaf(t2, s2w1[2], fmaf(t3, s2w1[3], s2b1))));
    const float s0 = fast_sigmoid(z0 - z1);
    const float s1 = 1.0f - s0;

    // features per lane (row = base + lane): f7 = 0 pad
    const float f0 = x.x, f1 = x.y, f2 = x.z, f3 = x.w, f4 = fconv, f5 = s0, f6 = s1;

    // ================= tile 0: rows base..base+15 (feature source lanes 0-15) ======
    v8f c1 = {};
    c1 = wmma4(a1c0, mkv2(bt0(f0, f2), bt0(f1, f3)), c1);          // K = 0..3
    c1 = wmma4(a1c1, mkv2(bt0(f4, f6), lo ? f5 : 0.0f), c1);       // K = 4..7 (f7 = 0)
#pragma unroll
    for (int v = 0; v < 8; ++v) c1[v] = fast_tanh(c1[v] + eb1v[v]);
    v8f c2 = {};
    c2 = wmma4(a2c0, mkv2(bt0(c1[0], c1[2]), bt0(c1[1], c1[3])), c2);
    c2 = wmma4(a2c1, mkv2(bt0(c1[4], c1[6]), bt0(c1[5], c1[7])), c2);
    float r0 = eb3s;
    r0 = fmaf(fast_tanh(c2[0] + eb2v[0]), w3v[0], r0);
    r0 = fmaf(fast_tanh(c2[1] + eb2v[1]), w3v[1], r0);
    r0 = fmaf(fast_tanh(c2[2] + eb2v[2]), w3v[2], r0);
    r0 = fmaf(fast_tanh(c2[3] + eb2v[3]), w3v[3], r0);             // valid in lanes 0-15

    // ================= tile 1: rows base+16..base+31 (feature source lanes 16-31) ==
    v8f d1 = {};
    d1 = wmma4(a1c0, mkv2(bt1(f0, f2), bt1(f1, f3)), d1);
    d1 = wmma4(a1c1, mkv2(bt1(f4, f6), bt1(f5, 0.0f)), d1);
#pragma unroll
    for (int v = 0; v < 8; ++v) d1[v] = fast_tanh(d1[v] + eb1v[v]);
    v8f d2 = {};
    d2 = wmma4(a2c0, mkv2(bt0(d1[0], d1[2]), bt0(d1[1], d1[3])), d2);  // C data lives in lanes 0-15
    d2 = wmma4(a2c1, mkv2(bt0(d1[4], d1[6]), bt0(d1[5], d1[7])), d2);
    float r1 = eb3s;
    r1 = fmaf(fast_tanh(d2[0] + eb2v[0]), w3v[0], r1);
    r1 = fmaf(fast_tanh(d2[1] + eb2v[1]), w3v[1], r1);
    r1 = fmaf(fast_tanh(d2[2] + eb2v[2]), w3v[2], r1);
    r1 = fmaf(fast_tanh(d2[3] + eb2v[3]), w3v[3], r1);             // valid in lanes 0-15

    // merge the two tiles into one coalesced 128B nontemporal store
    float tmerge = half_swap(r1);
    __builtin_nontemporal_store(lo ? r0 : tmerge, &out[base + lane]);
  }

  // ---- scalar tail (rows not in a full 32-group; empty for B = 4M) ----
  const int tailStart = groups << 5;
  const int gtid = blockIdx.x * blockDim.x + threadIdx.x;
  const int gstride = gridDim.x * blockDim.x;
  for (int r = tailStart + gtid; r < n; r += gstride) {
    const v4f x = in4[r];
    float cacc = fmaf(x.x, cw0, fmaf(x.y, cw1, fmaf(x.z, cw2, fmaf(x.w, cw3, cb))));
    float fc = fast_sigmoid(cacc);
    float u0 = fast_tanh(fmaf(x.x, s1w0[0], fmaf(x.y, s1w1[0], s1b[0])));
    float u1 = fast_tanh(fmaf(x.x, s1w0[1], fmaf(x.y, s1w1[1], s1b[1])));
    float u2 = fast_tanh(fmaf(x.x, s1w0[2], fmaf(x.y, s1w1[2], s1b[2])));
    float u3 = fast_tanh(fmaf(x.x, s1w0[3], fmaf(x.y, s1w1[3], s1b[3])));
    float z0 = fmaf(u0, s2w0[0], fmaf(u1, s2w0[1], fmaf(u2, s2w0[2], fmaf(u3, s2w0[3], s2b0))));
    float z1 = fmaf(u0, s2w1[0], fmaf(u1, s2w1[1], fmaf(u2, s2w1[2], fmaf(u3, s2w1[3], s2b1))));
    float p0 = fast_sigmoid(z0 - z1);
    float f[7] = {x.x, x.y, x.z, x.w, fc, p0, 1.0f - p0};
    float h1[8];
#pragma unroll
    for (int o = 0; o < 8; ++o) {
      float acc = eb1v[o];
#pragma unroll
      for (int k = 0; k < 7; ++k) acc = fmaf(f[k], eW1[k * 8 + o], acc);
      h1[o] = fast_tanh(acc);
    }
    float h2[4];
#pragma unroll
    for (int o = 0; o < 4; ++o) {
      float acc = eb2v[o];
#pragma unroll
      for (int k = 0; k < 8; ++k) acc = fmaf(h1[k], eW2[k * 4 + o], acc);
      h2[o] = fast_tanh(acc);
    }
    float res = eb3s;
#pragma unroll
    for (int k = 0; k < 4; ++k) res = fmaf(h2[k], w3v[k], res);
    out[r] = res;
  }
}

extern "C" void kernel_launch(void* const* d_in, const int* in_sizes, int n_in,
                              void* d_out, int out_size, void* d_ws, size_t ws_size,
                              hipStream_t stream) {
  const float* in     = (const float*)d_in[0];
  const float* conv_w = (const float*)d_in[1];
  const float* conv_b = (const float*)d_in[2];
  const float* sW1    = (const float*)d_in[3];
  const float* sb1    = (const float*)d_in[4];
  const float* sW2    = (const float*)d_in[5];
  const float* sb2    = (const float*)d_in[6];
  const float* eW1    = (const float*)d_in[7];
  const float* eb1    = (const float*)d_in[8];
  const float* eW2    = (const float*)d_in[9];
  const float* eb2    = (const float*)d_in[10];
  const float* eW3    = (const float*)d_in[11];
  const float* eb3    = (const float*)d_in[12];
  float* out = (float*)d_out;

  const int n = in_sizes[0] / 4;           // number of rows
  const int groups = n >> 5;               // 32 rows per wave-group
  int blocks = (groups + 7) / 8;           // 8 waves per 256-thread block
  if (blocks > 4096) blocks = 4096;        // grid-stride; amortize per-wave setup
  if (blocks < 1) blocks = 1;

  estimator_qnn_kernel<<<blocks, 256, 0, stream>>>(
      in, conv_w, conv_b, sW1, sb1, sW2, sb2,
      eW1, eb1, eW2, eb2, eW3, eb3, out, n);
}